// HungarianMatcher_87454124081436
// MI455X (gfx1250) — compile-verified
//
#include <hip/hip_runtime.h>
#include <hip/hip_bf16.h>

#define BS     64
#define Q      300
#define KCLS   91
#define TT     4096
#define TPER   64
#define BQ     (BS * Q)

#define RROWS   8
#define TILE_T  1024
#define THREADS 256
#define HTHREADS 256
#define INFF    1e30f

// ---------------- CDNA5 async-to-LDS helpers (ASYNCcnt path) ----------------
__device__ __forceinline__ unsigned lds_off(const void* p) {
    // generic shared address = SHARED_BASE | lds_offset; low 32 bits are the LDS byte offset
    return (unsigned)(unsigned long long)p;
}
__device__ __forceinline__ void async_b32(unsigned lds, const void* g) {
    asm volatile("global_load_async_to_lds_b32 %0, %1, off"
                 :: "v"(lds), "v"(g) : "memory");
}
__device__ __forceinline__ void async_b128(unsigned lds, const void* g) {
    asm volatile("global_load_async_to_lds_b128 %0, %1, off"
                 :: "v"(lds), "v"(g) : "memory");
}
__device__ __forceinline__ void wait_async0() {
    asm volatile("s_wait_asynccnt 0" ::: "memory");
}

// ---------------- Kernel 1: pairwise DETR matching cost matrix ----------------
// C[bq, t] = 5*L1 + 1*(-prob[bq, label[t]]) + 2*(-GIoU)
__global__ __launch_bounds__(THREADS)
void cost_matrix_kernel(const float* __restrict__ probs,   // [BQ, 91]
                        const float* __restrict__ pred,    // [BQ, 4] cxcywh
                        const float* __restrict__ tgtb,    // [TT, 4] cxcywh
                        const int*   __restrict__ tlab,    // [TT]
                        float*       __restrict__ C)       // [BQ, TT]
{
    __shared__ float4 s_box[TILE_T];          // 16 KB
    __shared__ int    s_lab[TILE_T];          //  4 KB
    __shared__ float  s_prob[RROWS * 96];     //  3 KB (rows padded to 96)
    __shared__ float4 s_pred[RROWS];

    const int tid  = threadIdx.x;
    const int t0   = blockIdx.x * TILE_T;
    const int row0 = blockIdx.y * RROWS;

    // ---- async stage: target boxes (b128), labels (b32), prob rows, pred boxes ----
    #pragma unroll
    for (int k = 0; k < 4; ++k) {
        int idx = k * THREADS + tid;
        async_b128(lds_off(&s_box[idx]), tgtb + (size_t)(t0 + idx) * 4);
    }
    #pragma unroll
    for (int k = 0; k < 4; ++k) {
        int idx = k * THREADS + tid;
        async_b32(lds_off(&s_lab[idx]), tlab + (t0 + idx));
    }
    for (int i = tid; i < RROWS * KCLS; i += THREADS) {
        int r = i / KCLS, c = i - r * KCLS;
        async_b32(lds_off(&s_prob[r * 96 + c]), probs + (size_t)row0 * KCLS + i);
    }
    if (tid < RROWS)
        async_b128(lds_off(&s_pred[tid]), pred + (size_t)(row0 + tid) * 4);

    wait_async0();
    __syncthreads();

    // ---- per-thread: 4 consecutive targets, reused across 8 query rows ----
    const int base = tid * 4;
    float4 tb[4]; int lab4[4];
    float tax[4], tay[4], tbx[4], tby[4], areab[4];
    #pragma unroll
    for (int k = 0; k < 4; ++k) {
        tb[k]   = s_box[base + k];
        lab4[k] = s_lab[base + k];
        tax[k]  = tb[k].x - 0.5f * tb[k].z;
        tay[k]  = tb[k].y - 0.5f * tb[k].w;
        tbx[k]  = tb[k].x + 0.5f * tb[k].z;
        tby[k]  = tb[k].y + 0.5f * tb[k].w;
        areab[k] = (tbx[k] - tax[k]) * (tby[k] - tay[k]);
    }

    #pragma unroll
    for (int r = 0; r < RROWS; ++r) {
        float4 pb = s_pred[r];
        float pax = pb.x - 0.5f * pb.z, pay = pb.y - 0.5f * pb.w;
        float pbx = pb.x + 0.5f * pb.z, pby = pb.y + 0.5f * pb.w;
        float areaa = (pbx - pax) * (pby - pay);
        float o[4];
        #pragma unroll
        for (int k = 0; k < 4; ++k) {
            float l1 = fabsf(pb.x - tb[k].x) + fabsf(pb.y - tb[k].y)
                     + fabsf(pb.z - tb[k].z) + fabsf(pb.w - tb[k].w);
            float iw = fminf(pbx, tbx[k]) - fmaxf(pax, tax[k]); iw = fmaxf(iw, 0.0f);
            float ih = fminf(pby, tby[k]) - fmaxf(pay, tay[k]); ih = fmaxf(ih, 0.0f);
            float inter = iw * ih;
            float uni   = areaa + areab[k] - inter;
            float iou   = inter / uni;
            float cw = fmaxf(pbx, tbx[k]) - fminf(pax, tax[k]); cw = fmaxf(cw, 0.0f);
            float ch = fmaxf(pby, tby[k]) - fminf(pay, tay[k]); ch = fmaxf(ch, 0.0f);
            float ac = cw * ch;
            float giou = iou - (ac - uni) / ac;
            float pc = s_prob[r * 96 + lab4[k]];
            o[k] = 5.0f * l1 - pc - 2.0f * giou;
        }
        float4 st = make_float4(o[0], o[1], o[2], o[3]);
        *reinterpret_cast<float4*>(&C[(size_t)(row0 + r) * TT + t0 + base]) = st;   // b128 store
    }
}

// ---------------- Kernel 2: per-batch Hungarian (JV) on the 64x300 sub-tile ----------------
__global__ __launch_bounds__(HTHREADS)
void hungarian_kernel(const float* __restrict__ C,       // [BQ, TT]
                      float* __restrict__ pred_idx,      // [BS, 64]
                      float* __restrict__ tgt_idx)       // [BS, 64]
{
    __shared__ float s_cost[TPER * Q];        // 75 KB: cost[t][q], t=target row, q=query col
    __shared__ float s_u[TPER + 1];
    __shared__ float s_v[Q + 1];
    __shared__ float s_minv[Q + 1];
    __shared__ int   s_p[Q + 1];
    __shared__ int   s_way[Q + 1];
    __shared__ int   s_used[Q + 1];
    __shared__ float red_v[HTHREADS];
    __shared__ int   red_i[HTHREADS];
    __shared__ int   s_j0, s_done;

    const int tid = threadIdx.x;
    const int b   = blockIdx.x;

    // stage transposed cost tile via async-to-LDS: s_cost[t*Q+q] = C[b, q, b*64+t]
    const float* cb = C + (size_t)b * Q * TT + (size_t)b * TPER;
    for (int idx = tid; idx < TPER * Q; idx += HTHREADS) {
        int q = idx >> 6;
        int t = idx & 63;
        async_b32(lds_off(&s_cost[t * Q + q]), cb + (size_t)q * TT + t);
    }
    for (int j = tid; j <= Q; j += HTHREADS) { s_v[j] = 0.0f; s_p[j] = 0; s_way[j] = 0; }
    if (tid <= TPER) s_u[tid] = 0.0f;
    wait_async0();
    __syncthreads();

    const int jA = tid + 1;     // columns 1..256
    const int jB = tid + 257;   // columns 257..300 (valid if <= Q)

    for (int i = 1; i <= TPER; ++i) {
        s_minv[jA] = INFF; s_used[jA] = 0;
        if (jB <= Q) { s_minv[jB] = INFF; s_used[jB] = 0; }
        if (tid == 0) { s_used[0] = 0; s_p[0] = i; s_j0 = 0; s_done = 0; }
        __syncthreads();

        while (true) {
            if (tid == 0) s_used[s_j0] = 1;
            __syncthreads();
            const int   j0  = s_j0;
            const int   i0  = s_p[j0];
            const float ui0 = s_u[i0];
            const float* crow = &s_cost[(i0 - 1) * Q];

            if (!s_used[jA]) {
                float cur = crow[jA - 1] - ui0 - s_v[jA];
                if (cur < s_minv[jA]) { s_minv[jA] = cur; s_way[jA] = j0; }
            }
            if (jB <= Q && !s_used[jB]) {
                float cur = crow[jB - 1] - ui0 - s_v[jB];
                if (cur < s_minv[jB]) { s_minv[jB] = cur; s_way[jB] = j0; }
            }
            // block-wide argmin over free columns (tie-break: lowest index, like np.argmin)
            float bv = s_used[jA] ? INFF : s_minv[jA];
            int   bj = jA;
            if (jB <= Q) {
                float v2 = s_used[jB] ? INFF : s_minv[jB];
                if (v2 < bv) { bv = v2; bj = jB; }
            }
            red_v[tid] = bv; red_i[tid] = bj;
            __syncthreads();
            for (int s = HTHREADS / 2; s > 0; s >>= 1) {
                if (tid < s) {
                    float ov = red_v[tid + s]; int oi = red_i[tid + s];
                    if (ov < red_v[tid] || (ov == red_v[tid] && oi < red_i[tid])) {
                        red_v[tid] = ov; red_i[tid] = oi;
                    }
                }
                __syncthreads();
            }
            const float delta = red_v[0];
            const int   j1    = red_i[0];

            // apply delta (used columns map to distinct rows -> race-free u updates)
            if (s_used[jA]) { s_v[jA] -= delta; s_u[s_p[jA]] += delta; }
            else            { s_minv[jA] -= delta; }
            if (jB <= Q) {
                if (s_used[jB]) { s_v[jB] -= delta; s_u[s_p[jB]] += delta; }
                else            { s_minv[jB] -= delta; }
            }
            if (tid == 0) s_u[s_p[0]] += delta;
            __syncthreads();
            if (tid == 0) { s_j0 = j1; s_done = (s_p[j1] == 0) ? 1 : 0; }
            __syncthreads();
            if (s_done) break;
        }
        // augmenting path rewrite
        if (tid == 0) {
            int j0 = s_j0;
            while (j0) {
                int j1 = s_way[j0];
                s_p[j0] = s_p[j1];
                j0 = j1;
            }
        }
        __syncthreads();
    }

    if (tid == 0) {
        int cnt = 0;
        for (int j = 1; j <= Q; ++j) {
            if (s_p[j] != 0) {
                pred_idx[b * TPER + cnt] = (float)(j - 1);
                tgt_idx [b * TPER + cnt] = (float)(s_p[j] - 1);
                if (++cnt == TPER) break;
            }
        }
    }
}

// ---------------- launch ----------------
extern "C" void kernel_launch(void* const* d_in, const int* in_sizes, int n_in,
                              void* d_out, int out_size, void* d_ws, size_t ws_size,
                              hipStream_t stream) {
    (void)in_sizes; (void)n_in; (void)out_size; (void)d_ws; (void)ws_size;
    const float* probs = (const float*)d_in[0];   // [64,300,91]
    const float* pred  = (const float*)d_in[1];   // [64,300,4]
    const float* tgtb  = (const float*)d_in[2];   // [4096,4]
    const int*   tlab  = (const int*)d_in[3];     // [4096]

    float* C        = (float*)d_out;                      // 64*300*4096 floats
    float* pred_idx = C + (size_t)BS * Q * TT;            // 64*64
    float* tgt_idx  = pred_idx + (size_t)BS * TPER;       // 64*64

    dim3 g1(TT / TILE_T, BQ / RROWS);                     // (4, 2400)
    cost_matrix_kernel<<<g1, THREADS, 0, stream>>>(probs, pred, tgtb, tlab, C);
    hungarian_kernel<<<BS, HTHREADS, 0, stream>>>(C, pred_idx, tgt_idx);
}